// gnnf_1683627180313
// MI455X (gfx1250) — compile-verified
//
#include <hip/hip_runtime.h>
#include <math.h>

typedef __attribute__((ext_vector_type(2))) float v2f;
typedef __attribute__((ext_vector_type(8))) float v8f;

#define PD   144           // padded feature dim (140 -> 144, 9 tiles of 16)
#define DD   140
#define FIN  74
#define FINP 76            // padded input feature dim (74 -> 76, K%4==0)
#define NN   512
#define BB   32
#define MTOT (BB * NN)     // 16384 rows in node space
#define NEGV (-9e15f)

// ---------------- fp32 WMMA 16x16x4 wrapper -------------------------------
__device__ __forceinline__ v8f wmma4(v2f a, v2f b, v8f c) {
  return __builtin_amdgcn_wmma_f32_16x16x4_f32(false, a, false, b, (short)0, c,
                                               false, false);
}

// ---------------- packing kernels (run once; remove all hot-loop guards) --
__global__ void __launch_bounds__(256) pack_wep(const float* __restrict__ We,
                                                float* __restrict__ Wep) {
  int i = blockIdx.x * 256 + threadIdx.x;          // [144][76]
  if (i >= PD * FINP) return;
  int d = i / FINP, f = i % FINP;
  Wep[i] = (d < DD && f < FIN) ? We[d * FIN + f] : 0.0f;
}

__global__ void __launch_bounds__(256) pack_xe(const float* __restrict__ c_hs,
                                               float* __restrict__ Xe) {
  int i = blockIdx.x * 256 + threadIdx.x;          // [16384][76]
  int r = i / FINP, f = i % FINP;
  Xe[i] = (f < FIN) ? c_hs[(size_t)r * FIN + f] : 0.0f;
}

// per-layer packs: Wwp[col=d][k=f] = Ww[d][f]; Atp[col=e][k=d] = A[d][e]; Wbp
__global__ void __launch_bounds__(256) pack_layer(
    const float* __restrict__ Ww, const float* __restrict__ A,
    const float* __restrict__ Wb, float* __restrict__ Wwp,
    float* __restrict__ Atp, float* __restrict__ Wbp) {
  int i = blockIdx.x * 256 + threadIdx.x;
  if (i < PD * PD) {
    int d = i / PD, f = i % PD;
    Wwp[i] = (d < DD && f < DD) ? Ww[d * DD + f] : 0.0f;
  } else if (i < 2 * PD * PD) {
    int j = i - PD * PD;
    int e = j / PD, d = j % PD;
    Atp[j] = (d < DD && e < DD) ? A[d * DD + e] : 0.0f;
  } else if (i < 2 * PD * PD + PD) {
    int d = i - 2 * PD * PD;
    Wbp[d] = (d < DD) ? Wb[d] : 0.0f;
  }
}

// ---------------- C[MTOTxPD] = A[MTOTxlda] @ Bt^T (+bias), opt. C^T -------
// Bt is packed [144][ldb] with K contiguous: b-pair = one b64 load.
__global__ void __launch_bounds__(256) gemm_node(
    const float* __restrict__ A, int lda, int nK4,
    const float* __restrict__ Bt, int ldb, const float* __restrict__ bias,
    float* __restrict__ C, float* __restrict__ hT, int writeT) {
  const int wave = threadIdx.x >> 5, lane = threadIdx.x & 31;
  const int tile = blockIdx.x * 8 + wave;          // 1024*9 tiles
  const int mt = tile / 9, nt = tile % 9;
  const int row0 = mt << 4, col0 = nt << 4;
  const int m = lane & 15, half = lane >> 4, ks = half << 1;
  const float* arow = A + (size_t)(row0 + m) * lda + ks;
  const float* brow = Bt + (size_t)(col0 + m) * ldb + ks;
  v8f acc = {};
#pragma unroll 4
  for (int k = 0; k < nK4; ++k) {
    v2f a = *(const v2f*)(arow + (k << 2));
    v2f b = *(const v2f*)(brow + (k << 2));
    acc = wmma4(a, b, acc);
  }
  const int col = col0 + m;
  const float bv = bias ? bias[col] : 0.0f;
  const int rbase = row0 + (half << 3);
#pragma unroll
  for (int v = 0; v < 8; ++v) {
    const int r = rbase + v;
    const float val = acc[v] + bv;
    C[(size_t)r * PD + col] = val;
    if (writeT)
      hT[((size_t)(r >> 9) * PD + col) * NN + (r & (NN - 1))] = val;
  }
}

// ---------------- e_sym = hA@h^T + h@hA^T  (per batch; 2 N-tiles/wave) ----
__global__ void __launch_bounds__(256) egemm(
    const float* __restrict__ h, const float* __restrict__ hA,
    float* __restrict__ e) {
  const int b = blockIdx.y;
  const int wave = threadIdx.x >> 5, lane = threadIdx.x & 31;
  const int tile = blockIdx.x * 8 + wave;          // 512 pair-tiles / batch
  const int mt = tile >> 4, np = tile & 15;
  const int row0 = mt << 4, col0 = np << 5;
  const int m = lane & 15, half = lane >> 4, ks = half << 1;
  const float* Hb  = h  + (size_t)b * NN * PD;
  const float* HAb = hA + (size_t)b * NN * PD;
  const float* a1p = HAb + (size_t)(row0 + m) * PD + ks;
  const float* a2p = Hb  + (size_t)(row0 + m) * PD + ks;
  const float* b1p = Hb  + (size_t)(col0 + m) * PD + ks;
  const float* b1q = Hb  + (size_t)(col0 + 16 + m) * PD + ks;
  const float* b2p = HAb + (size_t)(col0 + m) * PD + ks;
  const float* b2q = HAb + (size_t)(col0 + 16 + m) * PD + ks;
  v8f acc0 = {}, acc1 = {};
#pragma unroll 4
  for (int k = 0; k < PD / 4; ++k) {
    const int off = k << 2;
    v2f a1 = *(const v2f*)(a1p + off);
    v2f a2 = *(const v2f*)(a2p + off);
    acc0 = wmma4(a1, *(const v2f*)(b1p + off), acc0);
    acc0 = wmma4(a2, *(const v2f*)(b2p + off), acc0);
    acc1 = wmma4(a1, *(const v2f*)(b1q + off), acc1);
    acc1 = wmma4(a2, *(const v2f*)(b2q + off), acc1);
  }
  float* Cb = e + (size_t)b * NN * NN;
  const int col = col0 + m;
  const int rbase = row0 + (half << 3);
#pragma unroll
  for (int v = 0; v < 8; ++v) {
    Cb[(size_t)(rbase + v) * NN + col]      = acc0[v];
    Cb[(size_t)(rbase + v) * NN + col + 16] = acc1[v];
  }
}

// ---------------- masked softmax over axis=1 (columns), att = sm * adj ----
__global__ void __launch_bounds__(256) masked_softmax(
    const float* __restrict__ e, const float* __restrict__ adj1,
    const float* __restrict__ dist, const float* __restrict__ mu_p,
    const float* __restrict__ dev_p, float* __restrict__ att, int mode) {
  const int b = blockIdx.y;
  const int k = blockIdx.x * 256 + threadIdx.x;
  const float* Eb = e    + (size_t)b * NN * NN;
  const float* A1 = adj1 + (size_t)b * NN * NN;
  const float* A2 = dist + (size_t)b * NN * NN;
  const float mu = mu_p[0];
  const float inv_dev = 1.0f / dev_p[0];
  float mx = -3.0e38f;
  for (int j = 0; j < NN; ++j) {
    float adj = A1[(size_t)j * NN + k];
    if (mode) { float d = A2[(size_t)j * NN + k] - mu; adj += __expf(-d * d * inv_dev); }
    float v = (adj > 0.0f) ? Eb[(size_t)j * NN + k] : NEGV;
    mx = fmaxf(mx, v);
  }
  float s = 0.0f;
  for (int j = 0; j < NN; ++j) {
    float adj = A1[(size_t)j * NN + k];
    if (mode) { float d = A2[(size_t)j * NN + k] - mu; adj += __expf(-d * d * inv_dev); }
    float v = (adj > 0.0f) ? Eb[(size_t)j * NN + k] : NEGV;
    s += __expf(v - mx);
  }
  float* Ab = att + (size_t)b * NN * NN;
  for (int j = 0; j < NN; ++j) {
    float adj = A1[(size_t)j * NN + k];
    if (mode) { float d = A2[(size_t)j * NN + k] - mu; adj += __expf(-d * d * inv_dev); }
    float v = (adj > 0.0f) ? Eb[(size_t)j * NN + k] : NEGV;
    Ab[(size_t)j * NN + k] = (__expf(v - mx) / s) * adj;
  }
}

// ---------------- hp = relu(att @ h) via hT (per batch, 512 x 144) --------
__global__ void __launch_bounds__(256) hpgemm(
    const float* __restrict__ att, const float* __restrict__ hT,
    float* __restrict__ hp) {
  const int b = blockIdx.y;
  const int wave = threadIdx.x >> 5, lane = threadIdx.x & 31;
  const int tile = blockIdx.x * 8 + wave;          // 32*9 tiles / batch
  const int mt = tile / 9, nt = tile % 9;
  const int row0 = mt << 4, col0 = nt << 4;
  const int m = lane & 15, half = lane >> 4, ks = half << 1;
  const float* arow = att + (size_t)b * NN * NN + (size_t)(row0 + m) * NN + ks;
  const float* brow = hT + (size_t)b * PD * NN + (size_t)(col0 + m) * NN + ks;
  v8f acc = {};
#pragma unroll 4
  for (int k = 0; k < NN / 4; ++k) {
    v2f a = *(const v2f*)(arow + (k << 2));
    v2f bb = *(const v2f*)(brow + (k << 2));
    acc = wmma4(a, bb, acc);
  }
  float* Cb = hp + (size_t)b * NN * PD;
  const int col = col0 + m;
  const int rbase = row0 + (half << 3);
#pragma unroll
  for (int v = 0; v < 8; ++v)
    Cb[(size_t)(rbase + v) * PD + col] = fmaxf(acc[v], 0.0f);
}

// ---------------- gating: coeff=sigmoid([x,hp].gw+gb); out per mode -------
__global__ void __launch_bounds__(128) gate_out(
    const float* __restrict__ x, const float* __restrict__ hp,
    const float* __restrict__ gw, const float* __restrict__ gb,
    const float* __restrict__ o1, float* __restrict__ out, int mode) {
  const int idx = blockIdx.x;
  const float* xr = x  + (size_t)idx * PD;
  const float* hr = hp + (size_t)idx * PD;
  __shared__ float red[128];
  float p = 0.0f;
  for (int d = threadIdx.x; d < DD; d += 128)
    p += xr[d] * gw[d] + hr[d] * gw[DD + d];
  red[threadIdx.x] = p;
  __syncthreads();
  for (int st = 64; st > 0; st >>= 1) {
    if (threadIdx.x < st) red[threadIdx.x] += red[threadIdx.x + st];
    __syncthreads();
  }
  const float coeff = 1.0f / (1.0f + __expf(-(red[0] + gb[0])));
  for (int d = threadIdx.x; d < PD; d += 128) {
    float v = coeff * xr[d] + (1.0f - coeff) * hr[d];
    if (mode) v -= o1[(size_t)idx * PD + d];
    out[(size_t)idx * PD + d] = v;
  }
}

// ---------------- masked readout + 4-layer MLP (one block per batch) ------
__global__ void __launch_bounds__(256) readout_mlp(
    const float* __restrict__ x, const float* __restrict__ valid,
    const float* __restrict__ w0, const float* __restrict__ b0,
    const float* __restrict__ w1, const float* __restrict__ b1,
    const float* __restrict__ w2, const float* __restrict__ b2,
    const float* __restrict__ w3, const float* __restrict__ b3,
    float* __restrict__ out) {
  const int b = blockIdx.x;
  __shared__ float g[DD], t0[128], t1[128];
  const float* xb = x + (size_t)b * NN * PD;
  const float* vb = valid + (size_t)b * NN;
  for (int d = threadIdx.x; d < DD; d += 256) {
    float s = 0.0f;
    for (int n = 0; n < NN; ++n) s += xb[(size_t)n * PD + d] * vb[n];
    g[d] = s;
  }
  __syncthreads();
  const int j = threadIdx.x;
  if (j < 128) {
    float a = b0[j];
    for (int d = 0; d < DD; ++d) a += g[d] * w0[d * 128 + j];
    t0[j] = fmaxf(a, 0.0f);
  }
  __syncthreads();
  if (j < 128) {
    float a = b1[j];
    for (int d = 0; d < 128; ++d) a += t0[d] * w1[d * 128 + j];
    t1[j] = fmaxf(a, 0.0f);
  }
  __syncthreads();
  if (j < 128) {
    float a = b2[j];
    for (int d = 0; d < 128; ++d) a += t1[d] * w2[d * 128 + j];
    t0[j] = fmaxf(a, 0.0f);
  }
  __syncthreads();
  if (j == 0) {
    float a = b3[0];
    for (int d = 0; d < 128; ++d) a += t0[d] * w3[d];
    out[b] = 1.0f / (1.0f + __expf(-a));
  }
}

// --------------------------------------------------------------------------
extern "C" void kernel_launch(void* const* d_in, const int* in_sizes, int n_in,
                              void* d_out, int out_size, void* d_ws,
                              size_t ws_size, hipStream_t stream) {
  const float* c_hs  = (const float*)d_in[0];
  const float* adjs1 = (const float*)d_in[1];
  const float* adjs2 = (const float*)d_in[2];
  const float* valid = (const float*)d_in[3];
  const float* We    = (const float*)d_in[4];
  const float* Ww    = (const float*)d_in[5];
  const float* Wb    = (const float*)d_in[6];
  const float* Amat  = (const float*)d_in[7];
  const float* gw    = (const float*)d_in[8];
  const float* gb    = (const float*)d_in[9];
  const float* mu    = (const float*)d_in[10];
  const float* dev   = (const float*)d_in[11];
  const float* fw0   = (const float*)d_in[12];
  const float* fb0   = (const float*)d_in[13];
  const float* fw1   = (const float*)d_in[14];
  const float* fb1   = (const float*)d_in[15];
  const float* fw2   = (const float*)d_in[16];
  const float* fb2   = (const float*)d_in[17];
  const float* fw3   = (const float*)d_in[18];
  const float* fb3   = (const float*)d_in[19];
  float* out = (float*)d_out;

  float* ws = (float*)d_ws;
  const size_t SZX = (size_t)MTOT * PD;            // 2,359,296 floats
  const size_t SZE = (size_t)BB * NN * NN;         // 8,388,608 floats
  float* x   = ws;
  float* h   = ws + 1 * SZX;
  float* hA  = ws + 2 * SZX;
  float* hp  = ws + 3 * SZX;
  float* o1  = ws + 4 * SZX;
  float* hT  = ws + 5 * SZX;                       // [BB][PD][NN]
  float* e   = ws + 6 * SZX;
  float* att = e + SZE;
  float* Wep = att + SZE;                          // [PD][FINP]
  float* Wwp = Wep + PD * FINP;                    // [PD][PD]
  float* Atp = Wwp + PD * PD;                      // [PD][PD]
  float* Wbp = Atp + PD * PD;                      // [PD]
  float* Xe  = hp;                                 // alias: dead before hp use

  const dim3 blk256(256), blk128(128);
  const int GEMM_BLOCKS = (1024 * 9) / 8;          // 1152

  pack_wep<<<(PD * FINP + 255) / 256, blk256, 0, stream>>>(We, Wep);
  pack_xe<<<(MTOT * FINP) / 256, blk256, 0, stream>>>(c_hs, Xe);

  // x = c_hs @ We^T   (branch-free, padded K=76)
  gemm_node<<<GEMM_BLOCKS, blk256, 0, stream>>>(Xe, FINP, FINP / 4, Wep, FINP,
                                                nullptr, x, nullptr, 0);

  for (int k = 0; k < 4; ++k) {
    const float* Wwk = Ww + (size_t)k * DD * DD;
    const float* Wbk = Wb + (size_t)k * DD;
    const float* Ak  = Amat + (size_t)k * DD * DD;
    const float* gwk = gw + (size_t)k * 2 * DD;
    const float* gbk = gb + k;

    pack_layer<<<(2 * PD * PD + PD + 255) / 256, blk256, 0, stream>>>(
        Wwk, Ak, Wbk, Wwp, Atp, Wbp);

    // shared across both gates: h (+hT), hA, symmetric logits e
    gemm_node<<<GEMM_BLOCKS, blk256, 0, stream>>>(x, PD, PD / 4, Wwp, PD, Wbp,
                                                  h, hT, 1);
    gemm_node<<<GEMM_BLOCKS, blk256, 0, stream>>>(h, PD, PD / 4, Atp, PD,
                                                  nullptr, hA, nullptr, 0);
    egemm<<<dim3(64, BB), blk256, 0, stream>>>(h, hA, e);

    // gate 1: covalent adjacency
    masked_softmax<<<dim3(2, BB), blk256, 0, stream>>>(e, adjs1, adjs2, mu,
                                                       dev, att, 0);
    hpgemm<<<dim3(36, BB), blk256, 0, stream>>>(att, hT, hp);
    gate_out<<<MTOT, blk128, 0, stream>>>(x, hp, gwk, gbk, nullptr, o1, 0);

    // gate 2: RBF adjacency; x <- out2 - out1 (in place, row-local)
    masked_softmax<<<dim3(2, BB), blk256, 0, stream>>>(e, adjs1, adjs2, mu,
                                                       dev, att, 1);
    hpgemm<<<dim3(36, BB), blk256, 0, stream>>>(att, hT, hp);
    gate_out<<<MTOT, blk128, 0, stream>>>(x, hp, gwk, gbk, o1, x, 1);
  }

  readout_mlp<<<BB, blk256, 0, stream>>>(x, valid, fw0, fb0, fw1, fb1, fw2,
                                         fb2, fw3, fb3, out);
}